// ICELUT_7808250544313
// MI455X (gfx1250) — compile-verified
//
#include <hip/hip_runtime.h>
#include <hip/hip_bf16.h>

typedef __attribute__((ext_vector_type(2))) float v2f;
typedef __attribute__((ext_vector_type(8))) float v8f;

#define DIMLUT   33
#define DSQ      (33 * 33)          // 1089
#define DCUBE    (33 * 33 * 33)     // 35937
#define NUMLUT   20
#define BATCH    4
#define NTOT     (3 * DCUBE)        // 107811  (flattened N of the GEMM)
#define IMG_H    1080
#define IMG_W    1920
#define NPIX     (IMG_H * IMG_W)    // 2073600
#define FAKES_N  (BATCH * 3 * NPIX) // 24883200
#define D3LUT_N  (BATCH * NTOT)     // 431244
#define MDIFF    2090880u           // 20*3*33*33*32 (elements per diff direction)
#define RED_BLOCKS 512

// ---------------------------------------------------------------------------
// Kernel 1: D3LUT[b, :] = sum_n weights[b,n] * LUTs[n, :]  via WMMA f32 16x16x4
// One wave per 16-column tile of N. A = weights padded 4->16 rows, K=20 in 5
// chunks of 4. All padding via value masks (no EXEC divergence before WMMA).
// ---------------------------------------------------------------------------
__global__ __launch_bounds__(256) void d3lut_gemm_wmma(
    const float* __restrict__ W,   // [4, 20]
    const float* __restrict__ L,   // [20, 107811]
    float* __restrict__ D)         // [4, 107811]
{
    const int nTiles = (NTOT + 15) / 16;
    int wave = (int)((blockIdx.x * blockDim.x + threadIdx.x) >> 5); // wave-uniform
    if (wave >= nTiles) return;                                     // uniform branch

    int lane = threadIdx.x & 31;
    int m    = lane & 15;      // A row / B,D column in tile
    int hi   = lane >> 4;      // 0: K even pair low, 1: high pair
    int c0   = wave * 16;
    int n    = c0 + m;         // B/D column for this lane

    int   nc    = (n < NTOT) ? n : (NTOT - 1);
    float nmask = (n < NTOT) ? 1.0f : 0.0f;
    int   mc    = (m < BATCH) ? m : 0;
    float amask = (m < BATCH) ? 1.0f : 0.0f;

    v8f acc = {0.f, 0.f, 0.f, 0.f, 0.f, 0.f, 0.f, 0.f};

#pragma unroll
    for (int k0 = 0; k0 < NUMLUT; k0 += 4) {
        int ka = k0 + hi * 2;
        // A fragment: weights[m][ka], weights[m][ka+1] (zero-padded rows 4..15)
        v2f a;
        a.x = W[mc * NUMLUT + ka]     * amask;
        a.y = W[mc * NUMLUT + ka + 1] * amask;
        // B fragment: LUTs[ka][n], LUTs[ka+1][n] (zero-padded cols >= NTOT)
        v2f b;
        b.x = L[(size_t)ka       * NTOT + nc] * nmask;
        b.y = L[(size_t)(ka + 1) * NTOT + nc] * nmask;
        // 8 args: (neg_a, A, neg_b, B, c_mod, C, reuse_a, reuse_b)
        acc = __builtin_amdgcn_wmma_f32_16x16x4_f32(
            false, a, false, b, (short)0, acc, false, false);
    }

    // D: lane holds column n, rows M = v + hi*8 ; only M<4 (hi==0, v<4) real.
    if (hi == 0 && n < NTOT) {
#pragma unroll
        for (int v = 0; v < BATCH; ++v)
            D[(size_t)v * NTOT + n] = acc[v];
    }
}

// ---------------------------------------------------------------------------
// Kernel 2: per-pixel trilinear LUT apply + residual add (memory-bound).
// One thread per (batch, pixel); handles all 3 channels sharing weights/ids.
// ---------------------------------------------------------------------------
__global__ __launch_bounds__(256) void trilerp_kernel(
    const float* __restrict__ img,   // [4,3,H,W]
    const float* __restrict__ lut,   // [4,3,33,33,33]
    float* __restrict__ out)         // [4,3,H,W]
{
    size_t tid = (size_t)blockIdx.x * blockDim.x + threadIdx.x;
    if (tid >= (size_t)BATCH * NPIX) return;
    int b = (int)(tid / NPIX);
    int p = (int)(tid % NPIX);

    const float invbin = (float)(32.0 / 1.000001); // 1/binsize

    const float* ib = img + (size_t)b * 3 * NPIX;
    float r  = ib[p];
    float g  = ib[(size_t)NPIX + p];
    float bl = ib[2 * (size_t)NPIX + p];

    float fr = r * invbin, fg = g * invbin, fb = bl * invbin;
    float frf = floorf(fr), fgf = floorf(fg), fbf = floorf(fb);
    int rid = (int)frf, gid = (int)fgf, bid = (int)fbf;
    float rd = fr - frf, gd = fg - fgf, bd = fb - fbf;

    float rd1 = 1.f - rd, gd1 = 1.f - gd, bd1 = 1.f - bd;
    float w000 = rd1 * gd1 * bd1;
    float w100 = rd  * gd1 * bd1;
    float w010 = rd1 * gd  * bd1;
    float w110 = rd  * gd  * bd1;
    float w001 = rd1 * gd1 * bd;
    float w101 = rd  * gd1 * bd;
    float w011 = rd1 * gd  * bd;
    float w111 = rd  * gd  * bd;

    int base = bid * DSQ + gid * DIMLUT + rid;   // id = r + g*33 + b*1089
    const float* lb = lut + (size_t)b * 3 * DCUBE;
    float ov[3] = {r, g, bl};                    // residual add

#pragma unroll
    for (int c = 0; c < 3; ++c) {
        const float* lc = lb + (size_t)c * DCUBE + base;
        float v = w000 * lc[0]
                + w100 * lc[1]
                + w010 * lc[DIMLUT]
                + w110 * lc[DIMLUT + 1]
                + w001 * lc[DSQ]
                + w101 * lc[DSQ + 1]
                + w011 * lc[DSQ + DIMLUT]
                + w111 * lc[DSQ + DIMLUT + 1];
        out[((size_t)b * 3 + c) * NPIX + p] = v + ov[c];
    }
}

// ---------------------------------------------------------------------------
// Kernel 3a: TVMN partial sums (deterministic block reduction into d_ws).
// dir 0: diff along k (stride 1), dir 1: along j (stride 33), dir 2: along i
// (stride 1089). Edge diffs (index 0 or 31) weighted 2x.
// ---------------------------------------------------------------------------
__global__ __launch_bounds__(256) void tvmn_partial(
    const float* __restrict__ L,     // [20,3,33,33,33]
    float* __restrict__ part)        // [2 * RED_BLOCKS]
{
    float s0 = 0.f, s1 = 0.f;
    const unsigned total = 3u * MDIFF;
    unsigned stride = gridDim.x * blockDim.x;
    for (unsigned idx = blockIdx.x * blockDim.x + threadIdx.x;
         idx < total; idx += stride) {
        unsigned dir = idx / MDIFF;
        unsigned rem = idx % MDIFF;
        float dif, w;
        if (dir == 0) {
            unsigned k = rem % 32u;  unsigned q = rem / 32u;
            unsigned j = q % 33u;    q /= 33u;
            unsigned i = q % 33u;    q /= 33u;        // q = n*3+c
            unsigned base = ((q * 33u + i) * 33u + j) * 33u + k;
            dif = L[base] - L[base + 1];
            w = (k == 0u || k == 31u) ? 2.f : 1.f;
        } else if (dir == 1) {
            unsigned k = rem % 33u;  unsigned q = rem / 33u;
            unsigned j = q % 32u;    q /= 32u;
            unsigned i = q % 33u;    q /= 33u;
            unsigned base = ((q * 33u + i) * 33u + j) * 33u + k;
            dif = L[base] - L[base + 33u];
            w = (j == 0u || j == 31u) ? 2.f : 1.f;
        } else {
            unsigned k = rem % 33u;  unsigned q = rem / 33u;
            unsigned j = q % 33u;    q /= 33u;
            unsigned i = q % 32u;    q /= 32u;
            unsigned base = ((q * 33u + i) * 33u + j) * 33u + k;
            dif = L[base] - L[base + 1089u];
            w = (i == 0u || i == 31u) ? 2.f : 1.f;
        }
        s0 += dif * dif * w;
        float dw = dif * w;
        s1 += (dw > 0.f) ? dw * dw : 0.f;
    }

    __shared__ float sh0[256];
    __shared__ float sh1[256];
    sh0[threadIdx.x] = s0;
    sh1[threadIdx.x] = s1;
    __syncthreads();
    for (int off = 128; off > 0; off >>= 1) {
        if ((int)threadIdx.x < off) {
            sh0[threadIdx.x] += sh0[threadIdx.x + off];
            sh1[threadIdx.x] += sh1[threadIdx.x + off];
        }
        __syncthreads();
    }
    if (threadIdx.x == 0) {
        part[blockIdx.x]              = sh0[0];
        part[RED_BLOCKS + blockIdx.x] = sh1[0];
    }
}

// Kernel 3b: finish reduction, write scalar loss.
__global__ __launch_bounds__(256) void tvmn_final(
    const float* __restrict__ part, float* __restrict__ loss)
{
    __shared__ float sh0[256];
    __shared__ float sh1[256];
    float s0 = 0.f, s1 = 0.f;
    for (int i = threadIdx.x; i < RED_BLOCKS; i += 256) {
        s0 += part[i];
        s1 += part[RED_BLOCKS + i];
    }
    sh0[threadIdx.x] = s0;
    sh1[threadIdx.x] = s1;
    __syncthreads();
    for (int off = 128; off > 0; off >>= 1) {
        if ((int)threadIdx.x < off) {
            sh0[threadIdx.x] += sh0[threadIdx.x + off];
            sh1[threadIdx.x] += sh1[threadIdx.x + off];
        }
        __syncthreads();
    }
    if (threadIdx.x == 0) {
        const float invM = 1.0f / (float)MDIFF;
        loss[0] = 1e-4f * (sh0[0] * invM) + 10.0f * (sh1[0] * invM);
    }
}

extern "C" void kernel_launch(void* const* d_in, const int* in_sizes, int n_in,
                              void* d_out, int out_size, void* d_ws, size_t ws_size,
                              hipStream_t stream) {
    const float* weights = (const float*)d_in[0];  // [4,20]
    const float* LUTs    = (const float*)d_in[1];  // [20,3,33,33,33]
    const float* img     = (const float*)d_in[2];  // [4,3,1080,1920]

    float* out   = (float*)d_out;
    float* fakes = out;                       // 24,883,200
    float* d3lut = out + FAKES_N;             // 431,244
    float* loss  = out + FAKES_N + D3LUT_N;   // 1
    float* part  = (float*)d_ws;              // 2*RED_BLOCKS floats

    // 1) D3LUT via WMMA: one wave per 16-wide N tile (6739 waves, 8 waves/block)
    {
        int nTiles = (NTOT + 15) / 16;                 // 6739
        int blocks = (nTiles + 7) / 8;                 // 256 thr = 8 waves
        d3lut_gemm_wmma<<<blocks, 256, 0, stream>>>(weights, LUTs, d3lut);
    }

    // 2) Trilinear interpolation + residual add (depends on d3lut; same stream)
    {
        size_t tot = (size_t)BATCH * NPIX;
        int blocks = (int)((tot + 255) / 256);
        trilerp_kernel<<<blocks, 256, 0, stream>>>(img, d3lut, fakes);
    }

    // 3) TVMN regularizer (deterministic two-stage reduction)
    tvmn_partial<<<RED_BLOCKS, 256, 0, stream>>>(LUTs, part);
    tvmn_final<<<1, 256, 0, stream>>>(part, loss);
}